// GCN_NET_73332271612558
// MI455X (gfx1250) — compile-verified
//
#include <hip/hip_runtime.h>
#include <cmath>

// ---------------------------------------------------------------------------
// GCN 2-layer forward for MI455X (gfx1250, wave32).
//  - Dense GEMMs use V_WMMA_F32_16X16X4_F32 (native fp32 WMMA, matches the
//    fp32 reference precision; GEMM FLOPs are tiny vs. the scatter traffic).
//  - Edge aggregation (the memory-bound part, ~2 GB @ 23.3 TB/s) uses one
//    wave per edge with hardware global_atomic_add_f32 (unsafeAtomicAdd),
//    features coalesced across the 32 lanes.
// ---------------------------------------------------------------------------

typedef float v2f __attribute__((ext_vector_type(2)));
typedef float v8f __attribute__((ext_vector_type(8)));

#define GCN_FIN 128
#define GCN_HID 128
#define GCN_CLS 40

// ---------------- degree / normalization ----------------------------------

__global__ void gcn_fill_ones(float* __restrict__ p, int n) {
  int i = blockIdx.x * blockDim.x + threadIdx.x;
  if (i < n) p[i] = 1.0f;  // self-loop contributes 1 to every degree
}

__global__ void gcn_deg_accum(const int* __restrict__ row, float* __restrict__ deg, int E) {
  int e = blockIdx.x * blockDim.x + threadIdx.x;
  if (e < E) unsafeAtomicAdd(&deg[row[e]], 1.0f);
}

__global__ void gcn_deg_to_dinv(float* __restrict__ d, int n) {
  int i = blockIdx.x * blockDim.x + threadIdx.x;
  if (i < n) {
    float v = d[i];
    d[i] = (v > 0.0f) ? rsqrtf(v) : 0.0f;
  }
}

// ---------------- WMMA fp32 GEMM: C[M,N] = A[M,K] * B[K,N] -----------------
// One wave per 16x16 output tile. M must be a multiple of 16 (100000 = 6250*16).
// RAGGED=false: N is a multiple of 16 -> clean inner loop (loads + WMMA only).
// RAGGED=true (N=40): loads are address-clamped + value-selected so EXEC
// stays all-ones around the WMMA; only stores are predicated.

template <bool RAGGED>
__global__ void gcn_gemm_wmma_f32(const float* __restrict__ A,
                                  const float* __restrict__ B,
                                  float* __restrict__ C,
                                  int M, int K, int N, int ntilesN) {
  int wave = (blockIdx.x * blockDim.x + threadIdx.x) >> 5;  // uniform per wave
  int lane = threadIdx.x & 31;
  int tilesM = M >> 4;
  if (wave >= tilesM * ntilesN) return;  // wave-uniform exit

  int tm = wave / ntilesN;
  int tn = wave % ntilesN;

  int half = lane >> 4;   // 0: K+0/K+1, 1: K+2/K+3
  int l15  = lane & 15;
  int m = tm * 16 + l15;                       // A row for this lane
  int n = tn * 16 + l15;                       // B/C column for this lane
  bool nok = RAGGED ? (n < N) : true;
  int nc = RAGGED ? (nok ? n : 0) : n;         // clamped column for safe loads

  // Incrementing base pointers: A fragment is one aligned v2f (b64) load.
  const float* ap = A + (long)m * K + half * 2;       // advances by 4 per step
  const float* bp = B + (long)(half * 2) * N + nc;    // advances by 4*N per step
  const long bstep = (long)4 * N;

  v8f acc = {};
#pragma unroll 8
  for (int kb = 0; kb < K; kb += 4) {
    v2f a = *(const v2f*)ap;   // K even, half*2 even -> 8B aligned
    v2f b;
    b.x = bp[0];
    b.y = bp[N];
    if (RAGGED) {
      b.x = nok ? b.x : 0.0f;
      b.y = nok ? b.y : 0.0f;
    }
    acc = __builtin_amdgcn_wmma_f32_16x16x4_f32(
        /*neg_a=*/false, a, /*neg_b=*/false, b,
        /*c_mod=*/(short)0, acc, /*reuse_a=*/false, /*reuse_b=*/false);
    ap += 4;
    bp += bstep;
  }

  // C/D layout: VGPR v -> M = tm*16 + v + 8*half, N = tn*16 + (lane&15)
  int mbase = tm * 16 + half * 8;
  if (nok) {
    for (int v = 0; v < 8; ++v)
      C[(long)(mbase + v) * N + n] = acc[v];
  }
}

// ---------------- self-loop init: out[i,f] = H[i,f] * dinv[i]^2 ------------

__global__ void gcn_scale_self(const float* __restrict__ H,
                               const float* __restrict__ dinv,
                               float* __restrict__ out, int n, int F) {
  long i = (long)blockIdx.x * blockDim.x + threadIdx.x;
  if (i < (long)n * F) {
    int node = (int)(i / F);
    float w = dinv[node];
    out[i] = H[i] * w * w;
  }
}

// ---------------- edge aggregation: out[row] += H[col] * dinv[r]*dinv[c] ---
// One wave per edge: the F-wide feature row is coalesced across 32 lanes;
// each lane issues hardware float atomics (global_atomic_add_f32).

__global__ void gcn_propagate(const float* __restrict__ H,
                              const int* __restrict__ row,
                              const int* __restrict__ col,
                              const float* __restrict__ dinv,
                              float* __restrict__ out, int E, int F) {
  long gid = (long)blockIdx.x * blockDim.x + threadIdx.x;
  int e = (int)(gid >> 5);
  int lane = (int)(gid & 31);
  if (e >= E) return;
  int r = row[e];
  int c = col[e];
  float w = dinv[r] * dinv[c];
  const float* hc = H + (long)c * F;
  float* orow = out + (long)r * F;
  for (int f = lane; f < F; f += 32)
    unsafeAtomicAdd(&orow[f], hc[f] * w);
}

// ---------------- bias + ReLU (layer 1, F = 128) ---------------------------

__global__ void gcn_bias_relu(float* __restrict__ h, const float* __restrict__ b, long total) {
  long i = (long)blockIdx.x * blockDim.x + threadIdx.x;
  if (i < total) {
    float v = h[i] + b[i & (GCN_HID - 1)];
    h[i] = v > 0.0f ? v : 0.0f;
  }
}

// ---------------- bias + log_softmax over 40 classes, in place -------------

__global__ void gcn_bias_logsoftmax(float* __restrict__ out, const float* __restrict__ b, int n) {
  int i = blockIdx.x * blockDim.x + threadIdx.x;
  if (i >= n) return;
  float* p = out + (long)i * GCN_CLS;
  float v[GCN_CLS];
  float m = -3.402823466e38f;
  for (int c = 0; c < GCN_CLS; ++c) {
    v[c] = p[c] + b[c];
    m = fmaxf(m, v[c]);
  }
  float s = 0.0f;
  for (int c = 0; c < GCN_CLS; ++c) s += __expf(v[c] - m);
  float lse = m + __logf(s);
  for (int c = 0; c < GCN_CLS; ++c) p[c] = v[c] - lse;
}

// ---------------------------------------------------------------------------

extern "C" void kernel_launch(void* const* d_in, const int* in_sizes, int n_in,
                              void* d_out, int out_size, void* d_ws, size_t ws_size,
                              hipStream_t stream) {
  const float* x  = (const float*)d_in[0];
  const int*   ei = (const int*)d_in[1];   // [2, E] flat: row then col
  const float* W1 = (const float*)d_in[2];
  const float* b1 = (const float*)d_in[3];
  const float* W2 = (const float*)d_in[4];
  const float* b2 = (const float*)d_in[5];
  float* out = (float*)d_out;

  const int N = in_sizes[0] / GCN_FIN;   // 100000
  const int E = in_sizes[1] / 2;         // 1.6M

  const int* row = ei;
  const int* col = ei + E;

  // workspace layout (floats): dinv[N] | H1[N*128] | out1[N*128]; H2 reuses H1
  float* dinv = (float*)d_ws;
  float* H1   = dinv + N;
  float* out1 = H1 + (size_t)N * GCN_HID;
  float* H2   = H1;

  const int B = 256;

  // normalization: deg (with self loop) -> dinv
  gcn_fill_ones<<<(N + B - 1) / B, B, 0, stream>>>(dinv, N);
  gcn_deg_accum<<<(E + B - 1) / B, B, 0, stream>>>(row, dinv, E);
  gcn_deg_to_dinv<<<(N + B - 1) / B, B, 0, stream>>>(dinv, N);

  // layer 1: H1 = x @ W1  (WMMA, clean path), then aggregate, bias+ReLU
  {
    int ntn = GCN_HID / 16;
    long waves = (long)(N / 16) * ntn;
    gcn_gemm_wmma_f32<false><<<(unsigned)((waves * 32 + 127) / 128), 128, 0, stream>>>(
        x, W1, H1, N, GCN_FIN, GCN_HID, ntn);
  }
  {
    long total = (long)N * GCN_HID;
    gcn_scale_self<<<(unsigned)((total + B - 1) / B), B, 0, stream>>>(H1, dinv, out1, N, GCN_HID);
    gcn_propagate<<<(unsigned)(((long)E * 32 + B - 1) / B), B, 0, stream>>>(
        H1, row, col, dinv, out1, E, GCN_HID);
    gcn_bias_relu<<<(unsigned)((total + B - 1) / B), B, 0, stream>>>(out1, b1, total);
  }

  // layer 2: H2 = relu_h @ W2 (WMMA, ragged N=40), aggregate into d_out,
  // then bias + log_softmax in place
  {
    int ntn = (GCN_CLS + 15) / 16;  // 3
    long waves = (long)(N / 16) * ntn;
    gcn_gemm_wmma_f32<true><<<(unsigned)((waves * 32 + 127) / 128), 128, 0, stream>>>(
        out1, W2, H2, N, GCN_HID, GCN_CLS, ntn);
  }
  {
    long total = (long)N * GCN_CLS;
    gcn_scale_self<<<(unsigned)((total + B - 1) / B), B, 0, stream>>>(H2, dinv, out, N, GCN_CLS);
    gcn_propagate<<<(unsigned)(((long)E * 32 + B - 1) / B), B, 0, stream>>>(
        H2, row, col, dinv, out, E, GCN_CLS);
    gcn_bias_logsoftmax<<<(N + 127) / 128, 128, 0, stream>>>(out, b2, N);
  }
}